// Joint_Latent_opt_43095701848326
// MI455X (gfx1250) — compile-verified
//
#include <hip/hip_runtime.h>
#include <hip/hip_bf16.h>
#include <math.h>

#define D_FEAT 64
#define NEG_SLOPE 0.2f

typedef __attribute__((ext_vector_type(2))) float v2f;
typedef __attribute__((ext_vector_type(8))) float v8f;

// ---- order-preserving float <-> uint encoding (for exact atomic segment-max) ----
__device__ __forceinline__ unsigned ford(float f) {
  unsigned u = __float_as_uint(f);
  return (u & 0x80000000u) ? ~u : (u | 0x80000000u);
}
__device__ __forceinline__ float funord(unsigned u) {
  unsigned b = (u & 0x80000000u) ? (u & 0x7FFFFFFFu) : ~u;
  return __uint_as_float(b);
}

// ============================================================================
// K1: el = z @ attn_l, er = z @ attn_r via V_WMMA_F32_16X16X4_F32 (exact f32).
// One wave (32 lanes) computes a 16x16 C tile = Z[16x64] x B[64x16] where
// B col 0 = attn_l, col 1 = attn_r, cols 2..15 = 0. K walked in 16 steps of 4.
// Layouts per CDNA5 ISA 7.12.2:
//   A 16x4 f32 : lanes 0-15 -> M=lane, VGPR0/1 = K=0/1 ; lanes 16-31 -> K=2/3
//   B 4x16 f32 : lanes 0-15 -> N=lane, VGPR0/1 = K=0/1 ; lanes 16-31 -> K=2/3
//   C 16x16    : VGPR i, lanes 0-15 -> (M=i, N=lane); lanes 16-31 -> (M=8+i)
// B fragment is built branchlessly (pointer cndmask + multiplicative column
// mask) so the loop body has no EXEC manipulation around the WMMA.
// ============================================================================
__global__ void gat_el_er_wmma(const float* __restrict__ z,
                               const float* __restrict__ attn_l,
                               const float* __restrict__ attn_r,
                               float* __restrict__ el,
                               float* __restrict__ er,
                               int n_nodes) {
  const int lane = threadIdx.x;         // blockDim.x == 32
  const int base = blockIdx.x * 16;
  if (base >= n_nodes) return;          // uniform across the wave
  const int col = lane & 15;
  const int hi  = lane >> 4;            // 0 -> K pair {0,1}, 1 -> K pair {2,3}

  int row = base + col;
  if (row >= n_nodes) row = n_nodes - 1;   // clamp partial tile (stores guarded)
  const float* __restrict__ zr = z + (size_t)row * D_FEAT;

  // branchless B source: col 0 -> attn_l, col 1 -> attn_r, cols >= 2 -> zeroed
  const float* __restrict__ bp   = (col == 1) ? attn_r : attn_l;
  const float               mask = (col < 2) ? 1.f : 0.f;

  v8f c = {0.f, 0.f, 0.f, 0.f, 0.f, 0.f, 0.f, 0.f};
  #pragma unroll
  for (int kk = 0; kk < 16; ++kk) {
    const int k0 = kk * 4 + hi * 2;      // even -> 8B-aligned v2f loads
    v2f a = *(const v2f*)(zr + k0);
    v2f bv = *(const v2f*)(bp + k0);
    v2f b;
    b.x = bv.x * mask;
    b.y = bv.y * mask;
    c = __builtin_amdgcn_wmma_f32_16x16x4_f32(false, a, false, b,
                                              (short)0, c, false, false);
  }

  // el lives in C column 0 (lanes 0 and 16), er in column 1 (lanes 1 and 17).
  if (col < 2) {
    float* __restrict__ outp = (col == 0) ? el : er;
    const int m0 = base + hi * 8;
    #pragma unroll
    for (int i = 0; i < 8; ++i) {
      if (m0 + i < n_nodes) outp[m0 + i] = c[i];
    }
  }
}

// ============================================================================
// K2: per edge: e = leaky_relu(el[src] + er[dst]); segment-max into m_ord[dst]
// ============================================================================
__global__ void gat_edge_max(const float* __restrict__ el,
                             const float* __restrict__ er,
                             const int* __restrict__ src,
                             const int* __restrict__ dst,
                             float* __restrict__ e_buf,
                             unsigned* __restrict__ m_ord,
                             long n_edges) {
  long i = (long)blockIdx.x * blockDim.x + threadIdx.x;
  if (i >= n_edges) return;
  const int s = src[i];
  const int d = dst[i];
  float x = el[s] + er[d];
  float e = (x >= 0.f) ? x : NEG_SLOPE * x;
  e_buf[i] = e;
  atomicMax(&m_ord[d], ford(e));
}

// ============================================================================
// K3: per edge: ex = exp(e - m[dst]); segment-sum into den[dst]
// ============================================================================
__global__ void gat_edge_exp(const float* __restrict__ e_buf,
                             const int* __restrict__ dst,
                             const unsigned* __restrict__ m_ord,
                             float* __restrict__ ex_buf,
                             float* __restrict__ den,
                             long n_edges) {
  long i = (long)blockIdx.x * blockDim.x + threadIdx.x;
  if (i >= n_edges) return;
  const int d = dst[i];
  float ex = expf(e_buf[i] - funord(m_ord[d]));
  ex_buf[i] = ex;
  atomicAdd(&den[d], ex);
}

// ============================================================================
// K4: wave-per-edge weighted aggregation: rst[dst] += (ex/den[dst]) * z[src]
// Lane d handles features d and d+32 -> two coalesced 128B row loads and two
// coalesced bursts of global_atomic_add_f32 per edge. Everything is L2-resident
// (z = 12.8 MB << 192 MB), so this is L2 gather/atomic-throughput bound.
// ============================================================================
__global__ void gat_aggregate(const float* __restrict__ z,
                              const int* __restrict__ src,
                              const int* __restrict__ dst,
                              const float* __restrict__ ex_buf,
                              const float* __restrict__ den,
                              float* __restrict__ rst,
                              long n_edges) {
  const int wid  = threadIdx.x >> 5;    // 8 waves per 256-thread block
  const int lane = threadIdx.x & 31;
  long e = (long)blockIdx.x * 8 + wid;
  if (e >= n_edges) return;
  const int s = src[e];
  const int d = dst[e];
  const float a = ex_buf[e] / den[d];
  const float* __restrict__ zr = z + (size_t)s * D_FEAT;
  float* __restrict__ rr = rst + (size_t)d * D_FEAT;
  atomicAdd(rr + lane,      a * zr[lane]);
  atomicAdd(rr + lane + 32, a * zr[lane + 32]);
}

// ============================================================================
// K5: out = elu(rst)
// ============================================================================
__global__ void gat_elu(const float* __restrict__ rst,
                        float* __restrict__ out,
                        long n) {
  long i = (long)blockIdx.x * blockDim.x + threadIdx.x;
  if (i >= n) return;
  float x = rst[i];
  out[i] = (x > 0.f) ? x : expm1f(x);
}

extern "C" void kernel_launch(void* const* d_in, const int* in_sizes, int n_in,
                              void* d_out, int out_size, void* d_ws, size_t ws_size,
                              hipStream_t stream) {
  const float* z      = (const float*)d_in[0];
  const float* attn_l = (const float*)d_in[1];
  const float* attn_r = (const float*)d_in[2];
  const int*   src    = (const int*)d_in[3];
  const int*   dst    = (const int*)d_in[4];
  float*       out    = (float*)d_out;

  const int  n_nodes = in_sizes[0] / D_FEAT;
  const long n_edges = in_sizes[3];

  // ---- workspace carve-up (zeroed region first: rst | den | m_ord) ----
  char* ws = (char*)d_ws;
  float*    rst    = (float*)ws;    ws += (size_t)n_nodes * D_FEAT * sizeof(float);
  float*    den    = (float*)ws;    ws += (size_t)n_nodes * sizeof(float);
  unsigned* m_ord  = (unsigned*)ws; ws += (size_t)n_nodes * sizeof(unsigned);
  float*    el     = (float*)ws;    ws += (size_t)n_nodes * sizeof(float);
  float*    er     = (float*)ws;    ws += (size_t)n_nodes * sizeof(float);
  float*    e_buf  = (float*)ws;    ws += (size_t)n_edges * sizeof(float);
  float*    ex_buf = (float*)ws;    ws += (size_t)n_edges * sizeof(float);

  // zero accumulators: rst (N*64) + den (N) + m_ord (N; 0 encodes "-inf")
  const size_t zero_bytes = (size_t)n_nodes * (D_FEAT + 2) * sizeof(float);
  hipMemsetAsync(d_ws, 0, zero_bytes, stream);

  // K1: el/er via f32 WMMA, one wave per 16 nodes
  const int n_tiles = (n_nodes + 15) / 16;
  gat_el_er_wmma<<<n_tiles, 32, 0, stream>>>(z, attn_l, attn_r, el, er, n_nodes);

  // K2/K3: per-edge softmax passes
  const int  eb = 256;
  const long eg = (n_edges + eb - 1) / eb;
  gat_edge_max<<<(unsigned)eg, eb, 0, stream>>>(el, er, src, dst, e_buf, m_ord, n_edges);
  gat_edge_exp<<<(unsigned)eg, eb, 0, stream>>>(e_buf, dst, m_ord, ex_buf, den, n_edges);

  // K4: wave-per-edge aggregation (8 edges per 256-thread block)
  const long ag = (n_edges + 7) / 8;
  gat_aggregate<<<(unsigned)ag, 256, 0, stream>>>(z, src, dst, ex_buf, den, rst, n_edges);

  // K5: ELU over N*D
  const long nf = (long)n_nodes * D_FEAT;
  gat_elu<<<(unsigned)((nf + 255) / 256), 256, 0, stream>>>(rst, out, nf);
}